// parametricLoss_19945828122765
// MI455X (gfx1250) — compile-verified
//
#include <hip/hip_runtime.h>
#include <math.h>

#define N_BLOCKS  1024
#define N_THREADS 256
#define WAVE      32

typedef __attribute__((ext_vector_type(2))) float v2f;
typedef __attribute__((ext_vector_type(8))) float v8f;

// 32-point Gauss-Legendre nodes (positive half; full set is +/-x with same weights)
__device__ const float GLX[16] = {
  0.04830766568773832f, 0.14447196158279649f, 0.23928736225213707f, 0.33186860228212765f,
  0.42135127613063535f, 0.50689990893222939f, 0.58771575724076233f, 0.66304426693021520f,
  0.73218211874028968f, 0.79448379596794241f, 0.84936761373256997f, 0.89632115576605212f,
  0.93490607593773969f, 0.96476225558750643f, 0.98561151154526834f, 0.99726386184948156f };
__device__ const float GLW[16] = {
  0.09654008851472780f, 0.09563872007927486f, 0.09384439908080457f, 0.09117387869576388f,
  0.08765209300440381f, 0.08331192422694676f, 0.07819389578707031f, 0.07234579410884851f,
  0.06582222277636185f, 0.05868409347853555f, 0.05099805926237618f, 0.04283589802222668f,
  0.03427386291302143f, 0.02539206530926206f, 0.01627439473090567f, 0.00701861000947010f };

// Acklam inverse normal CDF (central branch is the only one hit for p in (0.05,0.95))
__device__ __forceinline__ float ndtri_f(float u) {
  const float plow = 0.02425f;
  float x;
  if (u < plow) {
    float q = sqrtf(-2.0f * logf(u));
    x = (((((-7.784894002430293e-03f*q - 3.223964580411365e-01f)*q - 2.400758277161838e+00f)*q
            - 2.549732539343734e+00f)*q + 4.374664141464968e+00f)*q + 2.938163982698783e+00f) /
        ((((7.784695709041462e-03f*q + 3.224671290700398e-01f)*q + 2.445134137142996e+00f)*q
            + 3.754408661907416e+00f)*q + 1.0f);
  } else if (u > 1.0f - plow) {
    float q = sqrtf(-2.0f * logf(1.0f - u));
    x = -(((((-7.784894002430293e-03f*q - 3.223964580411365e-01f)*q - 2.400758277161838e+00f)*q
            - 2.549732539343734e+00f)*q + 4.374664141464968e+00f)*q + 2.938163982698783e+00f) /
        ((((7.784695709041462e-03f*q + 3.224671290700398e-01f)*q + 2.445134137142996e+00f)*q
            + 3.754408661907416e+00f)*q + 1.0f);
  } else {
    float q = u - 0.5f;
    float r = q * q;
    x = (((((-3.969683028665376e+01f*r + 2.209460984245205e+02f)*r - 2.759285104469687e+02f)*r
            + 1.383577518672690e+02f)*r - 3.066479806614716e+01f)*r + 2.506628277459239e+00f) * q /
        (((((-5.447609879822406e+01f*r + 1.615858368580409e+02f)*r - 1.556989798598866e+02f)*r
            + 6.680131188771972e+01f)*r - 1.328068155288572e+01f)*r + 1.0f);
  }
  return x;
}

__device__ __forceinline__ float ndtr_f(float x) {
  return 0.5f * erfcf(-x * 0.7071067811865476f);
}

// Wave32 sum. WMMA path: A(16x4) with A.x = v (lanes 0..31 -> A[m][0], A[m][2]),
// A.y = 0; B(4x16) = ones  =>  D[m][n] = v[m] + v[m+16] for every column n.
// Lane n (<16) holds D rows 0..7, lane n+16 holds rows 8..15 -> one xor-16 add.
__device__ __forceinline__ float wave32_sum(float v) {
#if __has_builtin(__builtin_amdgcn_wmma_f32_16x16x4_f32)
  v2f a; a.x = v;    a.y = 0.0f;
  v2f b; b.x = 1.0f; b.y = 1.0f;
  v8f c = {};
  v8f d = __builtin_amdgcn_wmma_f32_16x16x4_f32(false, a, false, b, (short)0, c, false, false);
  float s = 0.0f;
  #pragma unroll
  for (int j = 0; j < 8; ++j) s += d[j];
  s += __shfl_xor(s, 16, 32);
  return s;
#else
  #pragma unroll
  for (int off = 16; off > 0; off >>= 1) v += __shfl_xor(v, off, 32);
  return v;
#endif
}

__global__ void __launch_bounds__(N_THREADS)
copula_loss_kernel(const float* __restrict__ yhat, const float* __restrict__ yv,
                   const float* __restrict__ g12,  const float* __restrict__ g34,
                   const float* __restrict__ g3412,const float* __restrict__ sig1,
                   const float* __restrict__ sig2, float* __restrict__ ws, int n) {
  __shared__ float4 s_nodes[32];                 // {alpha*log2e, beta*log2e, cw, 0}
  __shared__ float  s_wsum[N_THREADS / WAVE];

  // ---- uniform scalar precompute (tiny; all loads are cached/scalar) ----
  const float a0 = g12[0], a1 = g12[1], a2 = g12[2], a3 = g12[3];
  const float rd  = 1.0f / (a0 * a3 - a1 * a2);
  const float i00 = a3 * rd, i01 = -a1 * rd, i10 = -a2 * rd, i11 = a0 * rd;
  const float G00 = g3412[0], G01 = g3412[1], G10 = g3412[2], G11 = g3412[3];
  const float A00 = G00*i00 + G01*i10, A01 = G00*i01 + G01*i11;
  const float A10 = G10*i00 + G11*i10, A11 = G10*i01 + G11*i11;
  const float gs00 = g34[0] - (A00*G00 + A01*G01);
  const float gs01 = g34[1] - (A00*G10 + A01*G11);
  const float gs11 = g34[3] - (A10*G10 + A11*G11);
  const float s1g = sqrtf(gs00), s2g = sqrtf(gs11);
  const float rho = gs01 / (s1g * s2g);
  const float inv_s1 = 1.0f / sig1[0], inv_s2 = 1.0f / sig2[0];
  const float inv_s1g = 1.0f / s1g,    inv_s2g = 1.0f / s2g;
  const float qc = i01 + i10;

  if (threadIdx.x < 32) {
    const int j = threadIdx.x;
    const float xq = (j < 16) ? GLX[j] : -GLX[j - 16];
    const float wq = GLW[j & 15];
    const float r    = rho * 0.5f * (xq + 1.0f);
    const float omr  = 1.0f - r * r;
    const float inv2 = 0.5f / omr;
    const float L2E  = 1.4426950408889634f;
    const float alpha = inv2 * L2E;
    const float beta  = 2.0f * r * inv2 * L2E;
    const float cw = wq * (rho * 0.5f) * (1.0f / 6.283185307179586f) / sqrtf(omr);
    s_nodes[j] = make_float4(alpha, beta, cw, 0.0f);
  }
  __syncthreads();

  float partial = 0.0f;
  const int stride = gridDim.x * blockDim.x;
  for (int i = blockIdx.x * blockDim.x + threadIdx.x; i < n; i += stride) {
    const float p3  = yhat[i];
    const float m1p = yhat[i + n];
    const float p4  = yhat[i + 2 * n];
    const float m2p = yhat[i + 3 * n];
    const float l3  = yv[i];
    const float r1  = yv[i + n];
    const float l4  = yv[i + 2 * n];
    const float r2  = yv[i + 3 * n];

    const float e1 = (r1 - m1p) * inv_s1;
    const float e2 = (r2 - m2p) * inv_s2;
    const float quad = i00*e1*e1 + qc*e1*e2 + i11*e2*e2;
    const float mu1 = A00*e1 + A01*e2;
    const float mu2 = A10*e1 + A11*e2;

    const float u3 = fminf(fmaxf(1.0f - p3, 1e-7f), 1.0f - 1e-7f);
    const float u4 = fminf(fmaxf(1.0f - p4, 1e-7f), 1.0f - 1e-7f);
    const float h = (ndtri_f(u3) - mu1) * inv_s1g;
    const float k = (ndtri_f(u4) - mu2) * inv_s2g;

    const float pha = ndtr_f(h);
    const float phb = ndtr_f(k);

    // Drezner integral, node constants from LDS (ds_load_b128 broadcast)
    const float h2k2 = h * h + k * k;
    const float hk   = h * k;
    float I = 0.0f;
    #pragma unroll
    for (int j = 0; j < 32; ++j) {
      const float4 nd = s_nodes[j];
      const float t = fmaf(hk, nd.y, -h2k2 * nd.x);     // exponent, base-2 scaled
      I = fmaf(nd.z, __builtin_amdgcn_exp2f(t), I);      // v_exp_f32
    }

    // Corner algebra: Ci = Pa*Pb + sgn*I  (collapses 4 copula calls to 1)
    const bool b3 = (l3 >= 1.0f);
    const bool b4 = (l4 >= 1.0f);
    const float Pa  = b3 ? (1.0f - pha) : pha;
    const float Pb  = b4 ? (1.0f - phb) : phb;
    const float sgn = (b3 != b4) ? -1.0f : 1.0f;
    float Ci = fmaf(sgn, I, Pa * Pb);
    Ci = fmaxf(Ci, 1e-30f);
    const float logCi = 0.6931471805599453f * __builtin_amdgcn_logf(Ci); // v_log_f32
    partial += 0.5f * quad - logCi;
  }

  // deterministic block reduction: WMMA wave sum -> LDS -> one slot per block
  const float wsum = wave32_sum(partial);
  const int wave = threadIdx.x / WAVE;
  if ((threadIdx.x & (WAVE - 1)) == 0) s_wsum[wave] = wsum;
  __syncthreads();
  if (threadIdx.x == 0) {
    float bs = 0.0f;
    #pragma unroll
    for (int w = 0; w < N_THREADS / WAVE; ++w) bs += s_wsum[w];
    ws[blockIdx.x] = bs;
  }
}

__global__ void final_reduce_kernel(const float* __restrict__ ws,
                                    float* __restrict__ out, int m) {
  float s = 0.0f;
  for (int i = threadIdx.x; i < m; i += WAVE) s += ws[i];
  const float tot = wave32_sum(s);
  if (threadIdx.x == 0) out[0] = tot;
}

extern "C" void kernel_launch(void* const* d_in, const int* in_sizes, int n_in,
                              void* d_out, int out_size, void* d_ws, size_t ws_size,
                              hipStream_t stream) {
  const float* yhat  = (const float*)d_in[0];
  const float* yv    = (const float*)d_in[1];
  const float* g12   = (const float*)d_in[2];
  const float* g34   = (const float*)d_in[3];
  const float* g3412 = (const float*)d_in[4];
  const float* s1    = (const float*)d_in[5];
  const float* s2    = (const float*)d_in[6];
  const int n = in_sizes[0] / 4;           // y_hat is [4, N] flat
  float* ws = (float*)d_ws;                // N_BLOCKS floats of scratch

  copula_loss_kernel<<<N_BLOCKS, N_THREADS, 0, stream>>>(yhat, yv, g12, g34,
                                                         g3412, s1, s2, ws, n);
  final_reduce_kernel<<<1, WAVE, 0, stream>>>(ws, (float*)d_out, N_BLOCKS);
}